// FullQuantumTransformerVQC_12936441496149
// MI455X (gfx1250) — compile-verified
//
#include <hip/hip_runtime.h>

#define DEV __device__ __forceinline__

typedef float v2f __attribute__((ext_vector_type(2)));
typedef float v8f __attribute__((ext_vector_type(8)));

// ---------------------------------------------------------------------------
// 3-qubit register-resident statevector simulator (8 complex amplitudes).
// wire w (0 = MSB) <-> bit (2 - w); stride mask S = 4 >> w. All loops fully
// unrolled so the state stays in VGPRs.
// ---------------------------------------------------------------------------
DEV void g1q3(float sr[8], float si[8], int S,
              float m00r, float m00i, float m01r, float m01i,
              float m10r, float m10i, float m11r, float m11i) {
#pragma unroll
  for (int k = 0; k < 8; ++k) {
    if ((k & S) == 0) {
      float ar = sr[k], ai = si[k];
      float br = sr[k + S], bi = si[k + S];
      sr[k]     = m00r * ar - m00i * ai + m01r * br - m01i * bi;
      si[k]     = m00r * ai + m00i * ar + m01r * bi + m01i * br;
      sr[k + S] = m10r * ar - m10i * ai + m11r * br - m11i * bi;
      si[k + S] = m10r * ai + m10i * ar + m11r * bi + m11i * br;
    }
  }
}

DEV void cnot3(float sr[8], float si[8], int MC, int MT) {
#pragma unroll
  for (int k = 0; k < 8; ++k) {
    if ((k & MC) != 0 && (k & MT) == 0) {
      int j = k | MT;
      float tr = sr[k], ti = si[k];
      sr[k] = sr[j]; si[k] = si[j];
      sr[j] = tr;    si[j] = ti;
    }
  }
}

// w: flat [depth=4][3][3] (phi, theta, omega); AngleEmbedding RY + SEL.
DEV void sim3(const float ang[3], const float* __restrict__ w, float out[3], int nout) {
  float sr[8], si[8];
#pragma unroll
  for (int k = 0; k < 8; ++k) { sr[k] = (k == 0) ? 1.f : 0.f; si[k] = 0.f; }
#pragma unroll
  for (int q = 0; q < 3; ++q) {
    float a = ang[q] * 0.5f;
    float s = __sinf(a), c = __cosf(a);
    g1q3(sr, si, 4 >> q, c, 0.f, -s, 0.f, s, 0.f, c, 0.f);
  }
#pragma unroll
  for (int l = 0; l < 4; ++l) {
#pragma unroll
    for (int q = 0; q < 3; ++q) {
      float phi = w[l * 9 + q * 3 + 0];
      float th  = w[l * 9 + q * 3 + 1];
      float om  = w[l * 9 + q * 3 + 2];
      float ch = __cosf(th * 0.5f), sh = __sinf(th * 0.5f);
      float cp = __cosf((phi + om) * 0.5f), sp = __sinf((phi + om) * 0.5f);
      float cm = __cosf((phi - om) * 0.5f), sm = __sinf((phi - om) * 0.5f);
      // Rot(phi,theta,omega) = RZ(om) RY(th) RZ(phi)
      g1q3(sr, si, 4 >> q,
           cp * ch, -sp * ch,  -cm * sh, -sm * sh,
           cm * sh, -sm * sh,   cp * ch,  sp * ch);
    }
    int r = (l & 1) + 1;
#pragma unroll
    for (int q = 0; q < 3; ++q) {
      int t = q + r; if (t >= 3) t -= 3;
      cnot3(sr, si, 4 >> q, 4 >> t);
    }
  }
#pragma unroll
  for (int o = 0; o < 3; ++o) {
    if (o < nout) {
      int M = 4 >> o;
      float acc = 0.f;
#pragma unroll
      for (int k = 0; k < 8; ++k) {
        float p = sr[k] * sr[k] + si[k] * si[k];
        acc += (k & M) ? -p : p;
      }
      out[o] = acc;
    }
  }
}

// ---------------------------------------------------------------------------
// Stem: 10 tokens x 1024 batch, 3-qubit VQCs -> H (1024 x 30)
// ---------------------------------------------------------------------------
__global__ void __launch_bounds__(256) stem_k(const float* __restrict__ x,
                                              const float* __restrict__ th,
                                              float* __restrict__ H) {
  int tid = blockIdx.x * blockDim.x + threadIdx.x;
  if (tid >= 10 * 1024) return;
  int t = tid >> 10, b = tid & 1023;
  float ang[3];
#pragma unroll
  for (int j = 0; j < 3; ++j) ang[j] = x[b * 30 + t * 3 + j];
  float out[3];
  sim3(ang, th + t * 36, out, 3);
#pragma unroll
  for (int j = 0; j < 3; ++j) H[b * 30 + t * 3 + j] = out[j];
}

// ---------------------------------------------------------------------------
// FFN ('fully'): permuted gather, 3-qubit VQCs, residual add -> Hout
// ---------------------------------------------------------------------------
__global__ void __launch_bounds__(256) ffn_k(const float* __restrict__ Hin,
                                             const float* __restrict__ th,
                                             float* __restrict__ Hout) {
  int tid = blockIdx.x * blockDim.x + threadIdx.x;
  if (tid >= 10 * 1024) return;
  int k = tid >> 10, b = tid & 1023;
  float ang[3];
#pragma unroll
  for (int j = 0; j < 3; ++j) {
    int m = k * 3 + j;                      // flat idx of (3,10) row-major regrouped
    ang[j] = Hin[b * 30 + (m % 10) * 3 + (m / 10)];
  }
  float out[3];
  sim3(ang, th + k * 36, out, 3);
#pragma unroll
  for (int j = 0; j < 3; ++j)
    Hout[b * 30 + k * 3 + j] = Hin[b * 30 + k * 3 + j] + out[j];
}

// ---------------------------------------------------------------------------
// Reduce: 3->1 expval per token -> reduced (1024 x 12, cols 10/11 zeroed so
// the WMMA classifier can load A-fragments unconditionally)
// ---------------------------------------------------------------------------
__global__ void __launch_bounds__(256) reduce_k(const float* __restrict__ Hin,
                                                const float* __restrict__ th,
                                                float* __restrict__ red) {
  int tid = blockIdx.x * blockDim.x + threadIdx.x;
  if (tid >= 10 * 1024) return;
  int t = tid >> 10, b = tid & 1023;
  float ang[3];
#pragma unroll
  for (int j = 0; j < 3; ++j) ang[j] = Hin[b * 30 + t * 3 + j];
  float out[3];
  sim3(ang, th + t * 36, out, 1);
  red[b * 12 + t] = out[0];
  if (t == 0) { red[b * 12 + 10] = 0.f; red[b * 12 + 11] = 0.f; }
}

// ---------------------------------------------------------------------------
// 10-qubit attention circuit, one instance per 128-thread block, state in LDS
// (8 KB re/im). wire w <-> bit (9 - w). Gate matrices precomputed once per
// block into LDS (threads 0..49) to avoid redundant transcendentals.
// ---------------------------------------------------------------------------
DEV void gate1q_lds(float* re, float* im, int S, const float* m, int tid) {
  float m00r = m[0], m00i = m[1], m01r = m[2], m01i = m[3];
  float m10r = m[4], m10i = m[5], m11r = m[6], m11i = m[7];
  for (int p = tid; p < 512; p += 128) {
    int i0 = ((p & ~(S - 1)) << 1) | (p & (S - 1));
    int i1 = i0 + S;
    float ar = re[i0], ai = im[i0];
    float br = re[i1], bi = im[i1];
    re[i0] = m00r * ar - m00i * ai + m01r * br - m01i * bi;
    im[i0] = m00r * ai + m00i * ar + m01r * bi + m01i * br;
    re[i1] = m10r * ar - m10i * ai + m11r * br - m11i * bi;
    im[i1] = m10r * ai + m10i * ar + m11r * bi + m11i * br;
  }
  __syncthreads();
}

DEV void cnot_lds(float* re, float* im, int MC, int MT, int tid) {
  for (int i = tid; i < 1024; i += 128) {
    if ((i & MC) && !(i & MT)) {
      int j = i | MT;
      float tr = re[i], ti = im[i];
      re[i] = re[j]; im[i] = im[j];
      re[j] = tr;    im[j] = ti;
    }
  }
  __syncthreads();
}

__global__ void __launch_bounds__(128) attn_k(const float* __restrict__ Hin,
                                              const float* __restrict__ th_l, // theta_attn + l*720
                                              float* __restrict__ headout) {  // (1024 x 60)
  __shared__ float re[1024];
  __shared__ float im[1024];
  __shared__ float gm[50][8];   // 10 RY + 40 Rot gate matrices (complex 2x2)
  __shared__ float acc[10];
  int tid = threadIdx.x;
  int bid = blockIdx.x;
  int b = bid & 1023;
  int g = (bid >> 10) % 3;
  int h = bid / 3072;
  const float* w = th_l + (h * 3 + g) * 120;  // flat [4][10][3]

  for (int i = tid; i < 1024; i += 128) { re[i] = (i == 0) ? 1.f : 0.f; im[i] = 0.f; }
  if (tid < 10) {
    acc[tid] = 0.f;
    float a = Hin[b * 30 + tid * 3 + g] * 0.5f;
    float s = __sinf(a), c = __cosf(a);
    gm[tid][0] = c;  gm[tid][1] = 0.f;
    gm[tid][2] = -s; gm[tid][3] = 0.f;
    gm[tid][4] = s;  gm[tid][5] = 0.f;
    gm[tid][6] = c;  gm[tid][7] = 0.f;
  } else if (tid < 50) {
    int idx = tid - 10;                      // d*10 + q, matches weight layout
    float phi = w[idx * 3 + 0], tv = w[idx * 3 + 1], om = w[idx * 3 + 2];
    float ch = __cosf(tv * 0.5f), sh = __sinf(tv * 0.5f);
    float cp = __cosf((phi + om) * 0.5f), sp = __sinf((phi + om) * 0.5f);
    float cm = __cosf((phi - om) * 0.5f), sm = __sinf((phi - om) * 0.5f);
    gm[tid][0] = cp * ch;  gm[tid][1] = -sp * ch;
    gm[tid][2] = -cm * sh; gm[tid][3] = -sm * sh;
    gm[tid][4] = cm * sh;  gm[tid][5] = -sm * sh;
    gm[tid][6] = cp * ch;  gm[tid][7] = sp * ch;
  }
  __syncthreads();

  // AngleEmbedding RY
  for (int q = 0; q < 10; ++q)
    gate1q_lds(re, im, 1 << (9 - q), gm[q], tid);
  // StronglyEntanglingLayers, depth 4, range r = d+1
  for (int d = 0; d < 4; ++d) {
    for (int q = 0; q < 10; ++q)
      gate1q_lds(re, im, 1 << (9 - q), gm[10 + d * 10 + q], tid);
    int r = d + 1;
    for (int q = 0; q < 10; ++q) {
      int t = q + r; if (t >= 10) t -= 10;
      cnot_lds(re, im, 1 << (9 - q), 1 << (9 - t), tid);
    }
  }
  // <Z_t> for all 10 wires
  float a10[10];
#pragma unroll
  for (int t = 0; t < 10; ++t) a10[t] = 0.f;
  for (int i = tid; i < 1024; i += 128) {
    float p = re[i] * re[i] + im[i] * im[i];
#pragma unroll
    for (int t = 0; t < 10; ++t) a10[t] += (i & (1 << (9 - t))) ? -p : p;
  }
#pragma unroll
  for (int t = 0; t < 10; ++t) atomicAdd(&acc[t], a10[t]);
  __syncthreads();
  if (tid < 10) headout[b * 60 + h * 30 + tid * 3 + g] = acc[tid];
}

// ---------------------------------------------------------------------------
// WMMA fp32 GEMM: H(1024x30) += A(1024x60) @ W(30x60)^T, one 16x16 tile/wave.
// v_wmma_f32_16x16x4_f32, K stepped by 4 (15 steps). All loads unconditional
// (clamped addresses) + v_cndmask zeroing: no exec-mask branching.
// ---------------------------------------------------------------------------
__global__ void __launch_bounds__(32) gemm_wo_k(const float* __restrict__ A,
                                                const float* __restrict__ W,
                                                float* __restrict__ H) {
  int lane = threadIdx.x;
  int half = lane >> 4, lan = lane & 15;
  int m0 = blockIdx.x * 16;
  int n0 = blockIdx.y * 16;
  int nn = n0 + lan;
  bool nok = nn < 30;
  const float* arow = A + (m0 + lan) * 60;
  const float* brow = W + (nok ? nn : 0) * 60;   // clamped: always in-bounds
  v8f c = {};
#pragma unroll
  for (int kk = 0; kk < 15; ++kk) {
    int kb = kk * 4 + 2 * half;                  // even -> 8B aligned
    v2f a  = *(const v2f*)(arow + kb);
    v2f bv = *(const v2f*)(brow + kb);
    v2f bf;
    bf.x = nok ? bv.x : 0.f;
    bf.y = nok ? bv.y : 0.f;
    c = __builtin_amdgcn_wmma_f32_16x16x4_f32(false, a, false, bf,
                                              (short)0, c, false, false);
  }
#pragma unroll
  for (int r = 0; r < 8; ++r) {
    int m = m0 + r + 8 * half;
    if (nok) H[m * 30 + nn] += c[r];
  }
}

// ---------------------------------------------------------------------------
// WMMA classifier: out(1024x10) = reduced(1024x12) @ W_cls(10x10)^T + b_cls.
// W_cls staged into a zero-padded 16x12 LDS tile so all fragment loads are
// unconditional; K padded 10->12 (3 steps of 4), columns masked to 10.
// ---------------------------------------------------------------------------
__global__ void __launch_bounds__(32) cls_k(const float* __restrict__ R,
                                            const float* __restrict__ Wc,
                                            const float* __restrict__ bc,
                                            float* __restrict__ out) {
  __shared__ float wsh[16 * 12];
  int lane = threadIdx.x;
  int half = lane >> 4, lan = lane & 15;
  int m0 = blockIdx.x * 16;
  for (int i = lane; i < 16 * 12; i += 32) {
    int r = i / 12, k = i % 12;
    wsh[i] = (r < 10 && k < 10) ? Wc[r * 10 + k] : 0.f;
  }
  __syncthreads();
  const float* arow = R + (m0 + lan) * 12;
  v8f c = {};
#pragma unroll
  for (int kk = 0; kk < 3; ++kk) {
    int kb = kk * 4 + 2 * half;
    v2f a = *(const v2f*)(arow + kb);
    v2f bf;
    bf.x = wsh[lan * 12 + kb];
    bf.y = wsh[lan * 12 + kb + 1];
    c = __builtin_amdgcn_wmma_f32_16x16x4_f32(false, a, false, bf,
                                              (short)0, c, false, false);
  }
  if (lan < 10) {
    float bias = bc[lan];
#pragma unroll
    for (int r = 0; r < 8; ++r) {
      int m = m0 + r + 8 * half;
      out[m * 10 + lan] = c[r] + bias;
    }
  }
}

// ---------------------------------------------------------------------------
// Pipeline driver
// ---------------------------------------------------------------------------
extern "C" void kernel_launch(void* const* d_in, const int* in_sizes, int n_in,
                              void* d_out, int out_size, void* d_ws, size_t ws_size,
                              hipStream_t stream) {
  const float* x_      = (const float*)d_in[0];
  const float* th_stem = (const float*)d_in[1];
  const float* th_attn = (const float*)d_in[2];
  const float* W_O     = (const float*)d_in[3];
  const float* th_ffn  = (const float*)d_in[4];
  const float* th_red  = (const float*)d_in[5];
  const float* W_cls   = (const float*)d_in[6];
  const float* b_cls   = (const float*)d_in[7];
  float* out = (float*)d_out;

  float* ws  = (float*)d_ws;
  float* Ha  = ws;                      // 1024*30
  float* Hb  = ws + 30720;              // 1024*30
  float* HO  = ws + 61440;              // 1024*60 head outputs
  float* RED = ws + 61440 + 61440;      // 1024*12 (padded)

  stem_k<<<40, 256, 0, stream>>>(x_, th_stem, Ha);

  float* cur = Ha;
  float* nxt = Hb;
  for (int l = 0; l < 2; ++l) {
    attn_k<<<6144, 128, 0, stream>>>(cur, th_attn + l * 720, HO);
    dim3 g(64, 2);
    gemm_wo_k<<<g, 32, 0, stream>>>(HO, W_O + l * 1800, cur);
    ffn_k<<<40, 256, 0, stream>>>(cur, th_ffn + l * 360, nxt);
    float* t = cur; cur = nxt; nxt = t;
  }

  reduce_k<<<40, 256, 0, stream>>>(cur, th_red, RED);
  cls_k<<<64, 32, 0, stream>>>(RED, W_cls, b_cls, out);
}